// CostVolume_88175678587309
// MI455X (gfx1250) — compile-verified
//
#include <hip/hip_runtime.h>
#include <cstdint>
#include <cstddef>

typedef float v4f __attribute__((ext_vector_type(4)));

constexpr int kN = 8, kC = 64, kH = 160, kW = 320, kD = 53;
constexpr int kCH = 4;                 // channels staged per chunk
constexpr int kChunks = kC / kCH;      // 16 chunks
constexpr int kTW = 128;               // output columns per tile
constexpr int kTLW = kTW + 8;          // staged in1 width in dwords (halo of 4 each side)
constexpr int kXStep = 96;             // tile stride; tiles overlap, duplicate stores are identical

__device__ __forceinline__ uint32_t lds_lo32(const void* p) {
  // CDNA5 generic LDS pointers are {SHARED_BASE_hi32, lds_byte_offset}: low 32 bits are the LDS address.
  return (uint32_t)(uintptr_t)p;
}

__device__ __forceinline__ void async_copy_b128(uint32_t lds_byte, const void* sbase,
                                                uint32_t goff_byte) {
  // GVS mode: vdst = LDS byte address, vaddr = 32-bit byte offset, saddr = 64-bit base. ASYNCcnt-tracked.
  asm volatile("global_load_async_to_lds_b128 %0, %1, %2"
               :: "v"(lds_byte), "v"(goff_byte), "s"(sbase) : "memory");
}

__device__ __forceinline__ void wait_async_zero() {
  asm volatile("s_wait_asynccnt 0" ::: "memory");
}

// One c-chunk of accumulation for NR dy-rows sharing dx-pattern PAT.
template <int PAT, int NR, int NDX>
__device__ __forceinline__ void comp_step(const float* s1, const float* s2,
                                          const int* rows, int lane, v4f* acc) {
  #pragma unroll
  for (int cc = 0; cc < kCH; ++cc) {
    const v4f vv = *(const v4f*)(s2 + cc * kTW + 4 * lane);
    #pragma unroll
    for (int r = 0; r < NR; ++r) {
      const float* rp = s1 + (cc * 9 + rows[r]) * kTLW + 4 * lane;
      const v4f w0 = *(const v4f*)(rp);       // ds_load_b128
      const v4f w1 = *(const v4f*)(rp + 4);
      const v4f w2 = *(const v4f*)(rp + 8);
      float wl[12];
      *(v4f*)(&wl[0]) = w0; *(v4f*)(&wl[4]) = w1; *(v4f*)(&wl[8]) = w2;
      #pragma unroll
      for (int j = 0; j < NDX; ++j) {
        constexpr int dxt[4][7] = {
            {0, 2, 4, 6, 8, 0, 0},    // rows 0,8
            {1, 3, 5, 7, 0, 0, 0},    // rows 1,7
            {0, 2, 3, 4, 5, 6, 8},    // rows 2,4,6
            {1, 2, 3, 4, 5, 6, 7}};   // rows 3,5
        const int dx = dxt[PAT][j];
        v4f a;
        a.x = wl[dx + 0]; a.y = wl[dx + 1]; a.z = wl[dx + 2]; a.w = wl[dx + 3];
        acc[r * NDX + j] = __builtin_elementwise_fma(a, vv, acc[r * NDX + j]);
      }
    }
  }
}

template <int NR, int NDX>
__device__ __forceinline__ void store_acc(float* __restrict__ out, const v4f* acc,
                                          const int* rows, int n, int y, int xb) {
  constexpr int d0[9] = {0, 5, 9, 16, 23, 30, 37, 44, 48};  // first disp index of each dy row
  constexpr float s = 1.0f / 64.0f;
  #pragma unroll
  for (int r = 0; r < NR; ++r) {
    const int base = d0[rows[r]];
    #pragma unroll
    for (int j = 0; j < NDX; ++j) {
      v4f o = acc[r * NDX + j] * s;
      float* p = out + (((size_t)(n * kD + base + j) * kH + y) * kW + xb);
      __builtin_nontemporal_store(o, (v4f*)p);  // write-once output: keep L2 for inputs
    }
  }
}

__global__ __launch_bounds__(256) void corr53_kernel(const float* __restrict__ in1,
                                                     const float* __restrict__ in2,
                                                     float* __restrict__ out) {
  __shared__ __align__(16) float sIn1[2][kCH][9][kTLW];  // 39168 B, double-buffered
  __shared__ __align__(16) float sIn2[2][kCH][kTW];      //  4096 B

  const int tid = (int)threadIdx.x;
  const int lane = tid & 31;
  const int wv = tid >> 5;                // wave id 0..7 -> dy row owner
  const int x0 = (int)blockIdx.x * kXStep;
  const int y  = (int)blockIdx.y;
  const int n  = (int)blockIdx.z;

  const uint32_t lds1 = lds_lo32(&sIn1[0][0][0][0]);
  const uint32_t lds2 = lds_lo32(&sIn2[0][0][0]);

  // Zero in1 buffers once: halo / OOB packets are never overwritten, so they stay zero (pad).
  {
    v4f z = {0.f, 0.f, 0.f, 0.f};
    float* p = &sIn1[0][0][0][0];
    for (int i = tid; i < 2 * kCH * 9 * kTLW / 4; i += 256) ((v4f*)p)[i] = z;
  }
  __syncthreads();

  auto stage = [&](int chunk, int b) {
    const int c0 = chunk * kCH;
    // in2: kCH x 32 b128 packets, always in bounds (tile cols subset of [0,W))
    if (tid < kCH * 32) {
      const int cc = tid >> 5, k = tid & 31;
      uint32_t l = lds2 + (uint32_t)((((b * kCH + cc) * kTW) + 4 * k) * 4);
      uint32_t g = (uint32_t)(((((n * kC + c0 + cc) * kH) + y) * kW + x0 + 4 * k) * 4);
      async_copy_b128(l, in2, g);
    }
    // in1: kCH x 9 rows x 34 b128 packets, predicated per packet on row/col bounds
    for (int p = tid; p < kCH * 9 * 34; p += 256) {
      const int k = p % 34;
      const int t = p / 34;
      const int row = t % 9;
      const int cc = t / 9;
      const int col = x0 - 4 + 4 * k;   // 16B aligned (x0 % 4 == 0)
      const int yy = y + row - 4;
      if (yy >= 0 && yy < kH && col >= 0 && col <= kW - 4) {
        uint32_t l = lds1 + (uint32_t)(((((b * kCH + cc) * 9 + row) * kTLW) + 4 * k) * 4);
        uint32_t g = (uint32_t)(((((n * kC + c0 + cc) * kH) + yy) * kW + col) * 4);
        async_copy_b128(l, in1, g);
      }
    }
  };

  stage(0, 0);
  wait_async_zero();
  __syncthreads();

  v4f acc[10];  // worst case: wave 0 -> 2 rows x 5 dx
  #pragma unroll
  for (int i = 0; i < 10; ++i) { v4f z = {0.f, 0.f, 0.f, 0.f}; acc[i] = z; }

  int rows[2] = {wv, 8};  // wave 0 handles rows {0,8}; others handle row == wv

  for (int ch = 0; ch < kChunks; ++ch) {
    const int b = ch & 1;
    if (ch + 1 < kChunks) stage(ch + 1, b ^ 1);  // overlap next chunk's DMA with compute
    const float* s1 = &sIn1[b][0][0][0];
    const float* s2 = &sIn2[b][0][0];
    if (wv == 0)                   comp_step<0, 2, 5>(s1, s2, rows, lane, acc);
    else if (wv == 1 || wv == 7)   comp_step<1, 1, 4>(s1, s2, rows, lane, acc);
    else if (wv == 3 || wv == 5)   comp_step<3, 1, 7>(s1, s2, rows, lane, acc);
    else                           comp_step<2, 1, 7>(s1, s2, rows, lane, acc);
    wait_async_zero();   // my async copies (incl. just-issued next chunk) have landed in LDS
    __syncthreads();     // all waves' copies landed + all waves done reading previous buffer
  }

  const int xb = x0 + 4 * lane;
  if (wv == 0)                 store_acc<2, 5>(out, acc, rows, n, y, xb);
  else if (wv == 1 || wv == 7) store_acc<1, 4>(out, acc, rows, n, y, xb);
  else if (wv == 3 || wv == 5) store_acc<1, 7>(out, acc, rows, n, y, xb);
  else                         store_acc<1, 7>(out, acc, rows, n, y, xb);
}

extern "C" void kernel_launch(void* const* d_in, const int* in_sizes, int n_in,
                              void* d_out, int out_size, void* d_ws, size_t ws_size,
                              hipStream_t stream) {
  (void)in_sizes; (void)n_in; (void)out_size; (void)d_ws; (void)ws_size;
  const float* in1 = (const float*)d_in[0];
  const float* in2 = (const float*)d_in[1];
  float* out = (float*)d_out;
  dim3 grid((kW - kTW) / kXStep + 1, kH, kN);  // (3, 160, 8)
  dim3 block(256, 1, 1);
  corr53_kernel<<<grid, block, 0, stream>>>(in1, in2, out);
}